// SymmetrizedModel_83545703842168
// MI455X (gfx1250) — compile-verified
//
#include <hip/hip_runtime.h>

typedef __attribute__((ext_vector_type(2))) float v2f;
typedef __attribute__((ext_vector_type(8))) float v8f;

namespace {

constexpr int N_ROT = 256;   // contraction length K
constexpr int KC    = 128;   // K staged per LDS chunk == blockDim.x
constexpr int LSTR  = 132;   // LDS row stride in floats; 132 % 64 == 4 -> conflict-free b64 reads

// GEMM: OUT[M2, NCOLS] = sum_k (W[k]*D[k, r]) * V[k, col]
// A (weighted wigner) staged in LDS, B streamed once from global, C in WMMA accumulators.
template<int M2, int NT, int NCOLS>
__global__ __launch_bounds__(128) void wigner_proj_kernel(
    const float* __restrict__ W,   // [256] quadrature weights
    const float* __restrict__ D,   // [256, M2]  (mn contiguous)
    const float* __restrict__ V,   // [256, NCOLS] (i outer, s*c*p contiguous)
    float* __restrict__ OUT)       // [M2, NCOLS]
{
    constexpr int ROWS = NT * 16;            // padded row count (multiple of 16)
    __shared__ float Alds[ROWS * LSTR];      // <= 96*132*4 = 50688 B

    const int tid  = threadIdx.x;            // 0..127 (4 waves)
    const int wave = tid >> 5;
    const int lane = tid & 31;
    const int hh   = lane >> 4;              // half-wave selector
    const int l16  = lane & 15;

    const int col = (blockIdx.x * 4 + wave) * 16 + l16;

    const v8f vzero = {0.f,0.f,0.f,0.f,0.f,0.f,0.f,0.f};
    v8f acc[NT];
#pragma unroll
    for (int t = 0; t < NT; ++t) acc[t] = vzero;

    // B fragment: VGPR0 = row (k + 2*hh), VGPR1 = row (k + 2*hh + 1), col = l16
    const float* vp = V + (size_t)(2 * hh) * NCOLS + col;

    for (int kc = 0; kc < N_ROT; kc += KC) {
        if (kc) __syncthreads();             // previous chunk fully consumed

        // Stage Alds[r][k] = W[kc+k] * D[kc+k][r]; rows >= M2 zero-padded.
        // Thread tid owns column k = tid (KC == blockDim.x): compile-time
        // trip counts, no divergence, contiguous per-thread D reads,
        // conflict-free LDS stores (consecutive lanes -> consecutive banks).
        {
            const float wk = W[kc + tid];
            const float* __restrict__ dcol = D + (size_t)(kc + tid) * M2;
            float* __restrict__ lcol = Alds + tid;
#pragma unroll
            for (int r = 0; r < M2; ++r)
                lcol[r * LSTR] = wk * dcol[r];
#pragma unroll
            for (int r = M2; r < ROWS; ++r)
                lcol[r * LSTR] = 0.f;
        }
        __syncthreads();

#pragma unroll 4
        for (int k = 0; k < KC; k += 4) {
            v2f b;
            b.x = __builtin_nontemporal_load(vp);          // K = k+2*hh
            b.y = __builtin_nontemporal_load(vp + NCOLS);  // K = k+2*hh+1
            vp += (size_t)4 * NCOLS;
            const float* ap = &Alds[l16 * LSTR + k + 2 * hh];
#pragma unroll
            for (int t = 0; t < NT; ++t) {
                // A fragment: lane holds row (t*16 + l16), K = k+2*hh .. +1 (ds_load b64)
                v2f a = *(const v2f*)(ap + t * 16 * LSTR);
                acc[t] = __builtin_amdgcn_wmma_f32_16x16x4_f32(
                    /*neg_a=*/false, a, /*neg_b=*/false, b,
                    /*c_mod=*/(short)0, acc[t],
                    /*reuse_a=*/false, /*reuse_b=*/false);
            }
        }
    }

    // C tile layout: VGPR j -> row (j + 8*hh), lane -> column l16.
#pragma unroll
    for (int t = 0; t < NT; ++t) {
#pragma unroll
        for (int j = 0; j < 8; ++j) {
            int row = t * 16 + j + 8 * hh;
            if (row < M2) {
                __builtin_nontemporal_store(acc[t][j],
                    OUT + (size_t)row * NCOLS + col);
            }
        }
    }
}

template<int M2, int NT, int NCOLS>
inline void launch_ell(const float* W, const float* D, const float* V,
                       float* OUT, hipStream_t stream)
{
    dim3 grid(NCOLS / 64), block(128);       // 4 waves * 16 cols per block
    wigner_proj_kernel<M2, NT, NCOLS><<<grid, block, 0, stream>>>(W, D, V, OUT);
}

} // namespace

extern "C" void kernel_launch(void* const* d_in, const int* in_sizes, int n_in,
                              void* d_out, int out_size, void* d_ws, size_t ws_size,
                              hipStream_t stream)
{
    (void)in_sizes; (void)n_in; (void)d_ws; (void)ws_size; (void)out_size;

    // setup_inputs() dict order: weights, values_l0, D_l0, values_l1, D_l1, ...
    const float* W  = (const float*)d_in[0];
    const float* V0 = (const float*)d_in[1];  const float* D0 = (const float*)d_in[2];
    const float* V1 = (const float*)d_in[3];  const float* D1 = (const float*)d_in[4];
    const float* V2 = (const float*)d_in[5];  const float* D2 = (const float*)d_in[6];
    const float* V3 = (const float*)d_in[7];  const float* D3 = (const float*)d_in[8];
    const float* V4 = (const float*)d_in[9];  const float* D4 = (const float*)d_in[10];
    float* out = (float*)d_out;

    // NCOLS = 256 samples * m * 128 props = 32768 * m
    // Output offsets: cumulative M2 * NCOLS (fp32 elements).
    launch_ell< 1, 1,  32768>(W, D0, V0, out +        0, stream);
    launch_ell< 9, 1,  98304>(W, D1, V1, out +    32768, stream);
    launch_ell<25, 2, 163840>(W, D2, V2, out +   917504, stream);
    launch_ell<49, 4, 229376>(W, D3, V3, out +  5013504, stream);
    launch_ell<81, 6, 294912>(W, D4, V4, out + 16252928, stream);
}